// compute_G_11768210391385
// MI455X (gfx1250) — compile-verified
//
#include <hip/hip_runtime.h>

// ---------------------------------------------------------------------------
// G = DV2_H @ diag(W) @ invDE_HT_DV2   (all f32)
//   A = DV2_H        : [N, E] row-major  (cols scaled by W on LDS fill)
//   B = invDE_HT_DV2 : [E, N] row-major
//   C = G            : [N, N]
// N = 8192, E = 2048. Compute-bound -> V_WMMA_F32_16X16X4_F32.
// Block tile 128x256, 8 waves of 64x64 (4x4 WMMA tiles), KT=32,
// double-buffered LDS with register-staged global prefetch (1 barrier/iter).
// ---------------------------------------------------------------------------

typedef __attribute__((ext_vector_type(2))) float v2f;
typedef __attribute__((ext_vector_type(8))) float v8f;

#define N_DIM 8192
#define E_DIM 2048
#define KT    32          // K depth per LDS tile
#define NT    (E_DIM / KT)
#define LDA_P 36          // A tile pitch (floats): 16B rows, conflict-free col reads
#define LDB_P 260         // B tile pitch (floats): 16B rows
#define BM    128         // block rows
#define BN    256         // block cols

__launch_bounds__(256)
__global__ void hgnn_gemm_f32_wmma(const float* __restrict__ A,
                                   const float* __restrict__ B,
                                   const float* __restrict__ W,
                                   float* __restrict__ C) {
  __shared__ float sA[2][BM * LDA_P];   // 2 x 18432 B
  __shared__ float sB[2][KT * LDB_P];   // 2 x 33280 B

  const int tid  = threadIdx.x;
  const int lane = tid & 31;            // wave32
  const int wave = tid >> 5;            // 8 waves
  const int wm   = wave & 1;            // 2 waves along M (64 rows each)
  const int wn   = wave >> 1;           // 4 waves along N (64 cols each)

  const int bm = blockIdx.x * BM;
  const int bn = blockIdx.y * BN;

  const int lm = lane & 15;             // index within 16-wide tile edge
  const int kl = (lane >> 4) << 1;      // lanes 0-15 -> K{0,1}, lanes 16-31 -> K{2,3}

  // ---- global-fill thread mapping ----
  const int ar0 = tid >> 3;             // A: rows 0..31 (+32 per pass, 4 passes)
  const int ac4 = (tid & 7) << 2;       // A: col 0,4,...,28
  const int br0 = tid >> 6;             // B: rows 0..3  (+4 per pass, 8 passes)
  const int bc4 = (tid & 63) << 2;      // B: col 0..252

  const float* gA = A + (size_t)bm * E_DIM;
  const float* gB = B + bn;

  v8f acc[4][4];
  const v8f vzero = {0.f, 0.f, 0.f, 0.f, 0.f, 0.f, 0.f, 0.f};
#pragma unroll
  for (int i = 0; i < 4; ++i)
#pragma unroll
    for (int j = 0; j < 4; ++j)
      acc[i][j] = vzero;

  float4 aR[4];   // staged A tile fragment (scaled by W)
  float4 bR[8];   // staged B tile fragment

  // ---- register-stage one K-tile from global (A scaled by diag(W)) ----
  auto load_tile_regs = [&](int kt) {
    const float4 w4 = *reinterpret_cast<const float4*>(W + kt + ac4);
#pragma unroll
    for (int p = 0; p < 4; ++p) {
      const int row = ar0 + p * 32;
      float4 a4 = *reinterpret_cast<const float4*>(gA + (size_t)row * E_DIM + kt + ac4);
      a4.x *= w4.x; a4.y *= w4.y; a4.z *= w4.z; a4.w *= w4.w;
      aR[p] = a4;
    }
#pragma unroll
    for (int p = 0; p < 8; ++p) {
      const int row = br0 + p * 4;
      bR[p] = *reinterpret_cast<const float4*>(gB + (size_t)(kt + row) * N_DIM + bc4);
    }
  };

  auto store_tile_lds = [&](int buf) {
#pragma unroll
    for (int p = 0; p < 4; ++p)
      *reinterpret_cast<float4*>(&sA[buf][(ar0 + p * 32) * LDA_P + ac4]) = aR[p];
#pragma unroll
    for (int p = 0; p < 8; ++p)
      *reinterpret_cast<float4*>(&sB[buf][(br0 + p * 4) * LDB_P + bc4]) = bR[p];
  };

  // ---- prologue: tile 0 into buffer 0 ----
  load_tile_regs(0);
  store_tile_lds(0);
  __syncthreads();

  for (int t = 0; t < NT; ++t) {
    const int buf = t & 1;

    // issue next tile's global loads early; latency hidden under 128 WMMAs
    if (t + 1 < NT) load_tile_regs((t + 1) * KT);

    // ---- compute: 8 K-steps x 16 WMMAs from LDS buffer `buf` ----
#pragma unroll
    for (int k0 = 0; k0 < KT; k0 += 4) {
      v2f afrag[4];
#pragma unroll
      for (int i = 0; i < 4; ++i) {
        const int row = wm * 64 + i * 16 + lm;
        afrag[i] = *reinterpret_cast<const v2f*>(&sA[buf][row * LDA_P + k0 + kl]);
      }
      v2f bfrag[4];
#pragma unroll
      for (int j = 0; j < 4; ++j) {
        const int col = wn * 64 + j * 16 + lm;
        v2f b;
        b.x = sB[buf][(k0 + kl)     * LDB_P + col];
        b.y = sB[buf][(k0 + kl + 1) * LDB_P + col];
        bfrag[j] = b;
      }
#pragma unroll
      for (int i = 0; i < 4; ++i)
#pragma unroll
        for (int j = 0; j < 4; ++j)
          acc[i][j] = __builtin_amdgcn_wmma_f32_16x16x4_f32(
              /*neg_a=*/false, afrag[i],
              /*neg_b=*/false, bfrag[j],
              /*c_mod=*/(short)0, acc[i][j],
              /*reuse_a=*/false, /*reuse_b=*/false);
    }

    // ---- stage next tile into the other buffer; one barrier per iteration ----
    if (t + 1 < NT) store_tile_lds(buf ^ 1);
    __syncthreads();
  }

  // ---- store C: acc VGPR v holds M = v (lanes 0-15) / v+8 (lanes 16-31), N = lm ----
  const int half = lane >> 4;
#pragma unroll
  for (int i = 0; i < 4; ++i) {
#pragma unroll
    for (int j = 0; j < 4; ++j) {
      const int col = bn + wn * 64 + j * 16 + lm;
#pragma unroll
      for (int v = 0; v < 8; ++v) {
        const int row = bm + wm * 64 + i * 16 + half * 8 + v;
        C[(size_t)row * N_DIM + col] = acc[i][j][v];
      }
    }
  }
}

extern "C" void kernel_launch(void* const* d_in, const int* in_sizes, int n_in,
                              void* d_out, int out_size, void* d_ws, size_t ws_size,
                              hipStream_t stream) {
  const float* A = (const float*)d_in[0];  // DV2_H        [8192, 2048]
  const float* B = (const float*)d_in[1];  // invDE_HT_DV2 [2048, 8192]
  const float* W = (const float*)d_in[2];  // W            [2048]
  float* C = (float*)d_out;                // G            [8192, 8192]

  dim3 grid(N_DIM / BM, N_DIM / BN);       // 64 x 32 blocks
  hipLaunchKernelGGL(hgnn_gemm_f32_wmma, grid, dim3(256), 0, stream, A, B, W, C);
}